// TSA_PyTorch_34291018891929
// MI455X (gfx1250) — compile-verified
//
#include <hip/hip_runtime.h>
#include <math.h>
#include <stdint.h>

typedef __attribute__((ext_vector_type(2))) float v2f;
typedef __attribute__((ext_vector_type(8))) float v8f;
typedef unsigned int u32x4 __attribute__((ext_vector_type(4)));
typedef int          i32x4 __attribute__((ext_vector_type(4)));
typedef int          i32x8 __attribute__((ext_vector_type(8)));

// explicit global-address-space pointers (force global_load, not flat_load)
typedef __attribute__((address_space(1))) const v2f*   gv2f_cp;
typedef __attribute__((address_space(1))) const float* gf_cp;

#define ED    256
#define NH    8
#define NP    4
#define NBQ   2
#define HD    32
#define IMG_H 200
#define IMG_W 200
#define NQ    (IMG_H * IMG_W)
#define KC    64                 // K-chunk staged in LDS per TDM transfer
#define LDSB_BYTES (KC * 256 * 4)

// ---------------------------------------------------------------------------
// Tensor Data Mover: async 2D tile load Global -> LDS (D# per ISA ch.8).
// ---------------------------------------------------------------------------
__device__ __forceinline__ void tdm_load_tile_2d(unsigned lds_byte_off,
                                                 const float* gsrc,
                                                 unsigned tile_w, unsigned tile_h,
                                                 unsigned row_stride_elems)
{
  const unsigned long long ga = (unsigned long long)(uintptr_t)gsrc;
  u32x4 g0;
  g0[0] = 1u;                                   // count=1 (valid user D#)
  g0[1] = lds_byte_off;                         // lds_addr (bytes)
  g0[2] = (unsigned)ga;                         // global_addr[31:0]
  g0[3] = (unsigned)((ga >> 32) & 0x01FFFFFFu)  // global_addr[56:32]
        | (2u << 30);                           // type=2 ("image")
  i32x8 g1;
  g1[0] = (int)(2u << 16);                      // wg_mask=0, data_size=2 (4B)
  g1[1] = (int)((row_stride_elems & 0xFFFFu) << 16);          // tensor_dim0[15:0]
  g1[2] = (int)(((row_stride_elems >> 16) & 0xFFFFu)          // tensor_dim0[31:16]
              | ((tile_h & 0xFFFFu) << 16));                  // tensor_dim1[15:0]
  g1[3] = (int)((tile_w & 0xFFFFu) << 16);                    // tile_dim0
  g1[4] = (int)(tile_h & 0xFFFFu);                            // tile_dim1 (tile_dim2=0)
  g1[5] = (int)row_stride_elems;                              // tensor_dim0_stride[31:0]
  g1[6] = 0;
  g1[7] = 0;
  i32x4 gz = {0, 0, 0, 0};
#if defined(__clang_major__) && __clang_major__ >= 23
  i32x8 gz8 = {};
  __builtin_amdgcn_tensor_load_to_lds(g0, g1, gz, gz, gz8, 0);
#else
  __builtin_amdgcn_tensor_load_to_lds(g0, g1, gz, gz, 0);
#endif
}

// ---------------------------------------------------------------------------
// Wave-tile f32 WMMA GEMM, TDM double-buffered B panel in LDS:
//   C[M,N] = concat_K(A0,A1)[M,(1|2)K] @ B[(1|2)K,N] + bias[N] (+ residual)
// Block = 256 threads (8 waves) = one 16-row M block; waves split the N tiles
// (<=2 tiles/wave, sharing one A b64 load per lane). Wave 0 issues the TDM for
// chunk c+1 while all waves compute chunk c from the other LDS buffer.
// All guards are SGPR-scalar (readfirstlane) so EXEC stays all-ones at WMMA.
// M,N multiples of 16; K multiple of KC; N <= 256.
// ---------------------------------------------------------------------------
__global__ __launch_bounds__(256) void wmma_gemm_f32(
    const float* __restrict__ A0, const float* __restrict__ A1,
    const float* __restrict__ B,  const float* __restrict__ bias,
    const float* __restrict__ residual, float* __restrict__ C,
    int M, int N, int K)
{
  __shared__ float ldsB[2][KC * 256];                       // 2 x 64 KB buffers
  const unsigned ldsB_off = (unsigned)(uintptr_t)(&ldsB[0][0]);

  const int wave = __builtin_amdgcn_readfirstlane(threadIdx.x >> 5); // SGPR
  const int lane = threadIdx.x & 31;
  const int half = lane >> 4;
  const int l16  = lane & 15;
  const int tiles_n = N >> 4;

  const int tm   = blockIdx.x;                              // grid.x == M/16
  const int arow = tm * 16 + l16;

  // up to 2 N tiles per wave (all SGPR-uniform)
  int mytn[2] = {0, 0};
  int ntmine = 0;
  if (wave < tiles_n)     mytn[ntmine++] = wave;
  if (wave + 8 < tiles_n) mytn[ntmine++] = wave + 8;

  v8f acc[2] = {{}, {}};
  const int nparts = (A1 != nullptr) ? 2 : 1;
  const int chunks_per_part = K / KC;
  const int total_chunks = nparts * chunks_per_part;

  // prefetch chunk 0 into buffer 0
  if (wave == 0) tdm_load_tile_2d(ldsB_off, B, N, KC, N);

  int part = 0, k0 = 0;
  for (int c = 0; c < total_chunks; ++c) {
    const int buf = c & 1;
    if (wave == 0) __builtin_amdgcn_s_wait_tensorcnt(0);    // chunk c landed
    __syncthreads();                                        // publish buf to all

    // issue DMA for chunk c+1 into the other buffer (overlaps with compute)
    int npart = part, nk0 = k0 + KC;
    if (nk0 >= K) { nk0 = 0; npart = part + 1; }
    if (wave == 0 && c + 1 < total_chunks) {
      const float* Bn = B + ((size_t)npart * K + (size_t)nk0) * N;
      tdm_load_tile_2d(ldsB_off + (unsigned)(buf ^ 1) * LDSB_BYTES, Bn, N, KC, N);
    }

    const float* Ap = ((part == 0) ? A0 : A1) + (size_t)arow * K + k0;
    if (ntmine > 0) {
      for (int kk = 0; kk < KC; kk += 4) {
        const int ka = kk + 2 * half;       // lane K slot (ISA 32-bit A layout)
        v2f a = *(gv2f_cp)(uintptr_t)(Ap + ka);
#pragma unroll
        for (int t = 0; t < 2; ++t) {
          if (t < ntmine) {
            const int col = mytn[t] * 16 + l16;
            v2f b;
            b.x = ldsB[buf][(ka + 0) * N + col];
            b.y = ldsB[buf][(ka + 1) * N + col];
            acc[t] = __builtin_amdgcn_wmma_f32_16x16x4_f32(
                false, a, false, b, (short)0, acc[t], false, false);
          }
        }
      }
    }
    part = npart; k0 = nk0;
  }

  // D layout: VGPR i -> row tm*16 + i + 8*half
  for (int t = 0; t < ntmine; ++t) {
    const int col  = mytn[t] * 16 + l16;
    const float bval = bias ? bias[col] : 0.0f;
#pragma unroll
    for (int i = 0; i < 8; ++i) {
      const int r = tm * 16 + i + 8 * half;
      float v = acc[t][i] + bval;
      if (residual) v += *(gf_cp)(uintptr_t)(residual + (size_t)r * N + col);
      C[(size_t)r * N + col] = v;
    }
  }
}

// ---------------------------------------------------------------------------
// MSDA bilinear sampling + 4-point softmax + mean over the 2 temporal BEVs.
// One wave per (q, head); lane = head channel d -> each corner gather is one
// coalesced 128B load (value tensor is L2-resident).
// ---------------------------------------------------------------------------
__device__ __forceinline__ float bl_sample(const float* __restrict__ vbase,
                                           int ix, int iy) {
  const bool valid = ((unsigned)ix < (unsigned)IMG_W) &&
                     ((unsigned)iy < (unsigned)IMG_H);
  const int cx = ix < 0 ? 0 : (ix > IMG_W - 1 ? IMG_W - 1 : ix);
  const int cy = iy < 0 ? 0 : (iy > IMG_H - 1 ? IMG_H - 1 : iy);
  const float v = *(gf_cp)(uintptr_t)(vbase + ((size_t)cy * IMG_W + cx) * ED);
  return valid ? v : 0.0f;
}

__global__ __launch_bounds__(256) void msda_kernel(
    const float* __restrict__ value,    // [NBQ, NQ, NH*HD]
    const float* __restrict__ so_raw,   // [NQ, NH*NBQ*NP*2]
    const float* __restrict__ aw_raw,   // [NQ, NH*NBQ*NP]
    const float* __restrict__ ref_2d,   // [NBQ, NQ, 1, 2]
    float* __restrict__ msda_out)       // [NQ, ED]
{
  const int wid  = blockIdx.x * 8 + (threadIdx.x >> 5);
  const int lane = threadIdx.x & 31;
  const int q = wid >> 3;
  const int h = wid & 7;
  if (q >= NQ) return;

  float acc = 0.0f;
  for (int nb = 0; nb < NBQ; ++nb) {
    const float rx = ref_2d[((size_t)nb * NQ + q) * 2 + 0];
    const float ry = ref_2d[((size_t)nb * NQ + q) * 2 + 1];

    const float* awp = aw_raw + (size_t)q * (NH * NBQ * NP) + (h * NBQ + nb) * NP;
    const float a0 = awp[0], a1 = awp[1], a2 = awp[2], a3 = awp[3];
    const float m  = fmaxf(fmaxf(a0, a1), fmaxf(a2, a3));
    const float e0 = __expf(a0 - m), e1 = __expf(a1 - m);
    const float e2 = __expf(a2 - m), e3 = __expf(a3 - m);
    const float inv = 1.0f / (e0 + e1 + e2 + e3);
    const float w4[4] = {e0 * inv, e1 * inv, e2 * inv, e3 * inv};

    const float* sop   = so_raw + (size_t)q * (NH * NBQ * NP * 2) +
                         (size_t)(h * NBQ + nb) * NP * 2;
    const float* vbase = value + (size_t)nb * NQ * ED + h * HD + lane;

#pragma unroll
    for (int p = 0; p < NP; ++p) {
      const float lx = rx + sop[p * 2 + 0] * (1.0f / IMG_W);
      const float ly = ry + sop[p * 2 + 1] * (1.0f / IMG_H);
      const float xi = lx * (float)IMG_W - 0.5f;
      const float yi = ly * (float)IMG_H - 0.5f;
      const float x0f = floorf(xi), y0f = floorf(yi);
      const int   ix0 = (int)x0f,   iy0 = (int)y0f;
      const float wx = xi - x0f,    wy = yi - y0f;

      const float v00 = bl_sample(vbase, ix0,     iy0);
      const float v10 = bl_sample(vbase, ix0 + 1, iy0);
      const float v01 = bl_sample(vbase, ix0,     iy0 + 1);
      const float v11 = bl_sample(vbase, ix0 + 1, iy0 + 1);

      const float s = v00 * (1.0f - wx) * (1.0f - wy) +
                      v10 * wx          * (1.0f - wy) +
                      v01 * (1.0f - wx) * wy +
                      v11 * wx          * wy;
      acc += w4[p] * s;
    }
  }
  msda_out[(size_t)q * ED + h * HD + lane] = acc * (1.0f / NBQ);
}

// ---------------------------------------------------------------------------
extern "C" void kernel_launch(void* const* d_in, const int* in_sizes, int n_in,
                              void* d_out, int out_size, void* d_ws, size_t ws_size,
                              hipStream_t stream) {
  (void)in_sizes; (void)n_in; (void)out_size; (void)ws_size;
  const float* query    = (const float*)d_in[0];
  const float* prev_bev = (const float*)d_in[1];
  /* d_in[2] = bev_pos: unused by the reference */
  const float* ref_2d   = (const float*)d_in[3];
  const float* Wv       = (const float*)d_in[4];
  const float* bv       = (const float*)d_in[5];
  const float* Wso      = (const float*)d_in[6];
  const float* bso      = (const float*)d_in[7];
  const float* Waw      = (const float*)d_in[8];
  const float* baw      = (const float*)d_in[9];
  const float* Wo       = (const float*)d_in[10];
  const float* bo       = (const float*)d_in[11];
  /* d_in[12] = spatial_shapes: constants (200,200) hardcoded */
  float* out = (float*)d_out;

  float* ws     = (float*)d_ws;
  float* value  = ws;                                  // NBQ*NQ*ED
  float* so_raw = value  + (size_t)NBQ * NQ * ED;      // NQ*128
  float* aw_raw = so_raw + (size_t)NQ * (NH*NBQ*NP*2); // NQ*64
  float* msda   = aw_raw + (size_t)NQ * (NH*NBQ*NP);   // NQ*ED

  // 1) value = prev_bev @ Wv + bv            (M=80000, N=256, K=256)
  wmma_gemm_f32<<<(NBQ * NQ) / 16, 256, 0, stream>>>(
      prev_bev, nullptr, Wv, bv, nullptr, value, NBQ * NQ, ED, ED);

  // 2) so_raw = [prev_bev[0] | query] @ Wso + bso   (M=40000, N=128, K=512)
  wmma_gemm_f32<<<NQ / 16, 256, 0, stream>>>(
      prev_bev, query, Wso, bso, nullptr, so_raw, NQ, NH * NBQ * NP * 2, ED);

  // 3) aw_raw = [prev_bev[0] | query] @ Waw + baw   (M=40000, N=64, K=512)
  wmma_gemm_f32<<<NQ / 16, 256, 0, stream>>>(
      prev_bev, query, Waw, baw, nullptr, aw_raw, NQ, NH * NBQ * NP, ED);

  // 4) softmax + bilinear MSDA + mean over temporal BEVs
  msda_kernel<<<NQ, 256, 0, stream>>>(value, so_raw, aw_raw, ref_2d, msda);

  // 5) out = msda @ Wo + bo + query          (M=40000, N=256, K=256)
  wmma_gemm_f32<<<NQ / 16, 256, 0, stream>>>(
      msda, nullptr, Wo, bo, query, out, NQ, ED, ED);
}